// AffineTransform_40261023433399
// MI455X (gfx1250) — compile-verified
//
#include <hip/hip_runtime.h>

typedef __attribute__((ext_vector_type(2))) float v2f;
typedef __attribute__((ext_vector_type(4))) float v4f;
typedef __attribute__((ext_vector_type(8))) float v8f;

// Problem constants (from reference): B=4, H=512, W=512, D=32, C=1
#define NB 4
#define NH 512
#define NW 512
#define ND 32

__device__ __forceinline__ float wave_bcast(float v, int srcLane) {
    // Dst[lane] = src[srcLane]; byte index, wave32 uses bits [6:2] mod 32
    int r = __builtin_amdgcn_ds_bpermute(srcLane << 2, __float_as_int(v));
    return __int_as_float(r);
}

__global__ __launch_bounds__(256)
void affine_bilinear_kernel(const float* __restrict__ im,
                            const float* __restrict__ thetas,
                            float* __restrict__ out)
{
    const int lane  = threadIdx.x & 31;
    const int gtid  = blockIdx.x * 256 + threadIdx.x;
    const int pixel = gtid >> 3;      // 0 .. B*H*W-1 (8 lanes per pixel)
    const int chunk = gtid & 7;       // which float4 of the 32-float channel vector
    const int b     = pixel >> 18;    // pixel / (H*W); uniform across the wave
    const float* t  = thetas + b * 6; // [t00 t01 t02 t10 t11 t12]

    // ---------------- WMMA coordinate transform ----------------
    // new(2x3 @ 3xN) via V_WMMA_F32_16X16X4_F32.
    // A 16x4 layout: lanes0-15 -> M=lane, A.x=K0, A.y=K1; lanes16-31 -> M=lane-16, A.x=K2, A.y=K3
    v2f A;
    A.x = (lane == 0) ? t[0] : (lane == 1) ? t[3] :
          (lane == 16) ? t[2] : (lane == 17) ? t[5] : 0.0f;   // K0 rows / K2 rows
    A.y = (lane == 0) ? t[1] : (lane == 1) ? t[4] : 0.0f;     // K1 rows / K3 = 0

    // B 4x16 layout: B.x: lanes0-15 = row K0 (N=lane), lanes16-31 = row K2;
    //                B.y: lanes0-15 = row K1,          lanes16-31 = row K3.
    // Column n = [Xn, Yn, 1, 0]^T for pixel waveBase+n (only n<4 consumed).
    const int waveBase = (blockIdx.x * 256 + (threadIdx.x & ~31)) >> 3; // wave's first pixel
    const int n  = lane & 15;
    const int pn = waveBase + n;                   // arithmetic only; OOB cols unused
    const float sc = 2.0f / 511.0f;                // linspace(-1,1,512) step
    const float Xn = -1.0f + (float)(pn & 511) * sc;
    const float Yn = -1.0f + (float)((pn >> 9) & 511) * sc;
    v2f Bm;
    Bm.x = (lane < 16) ? Xn : 1.0f;                // K0 / K2(=ones row)
    Bm.y = (lane < 16) ? Yn : 0.0f;                // K1 / K3(=pad)

    v8f acc = {};
    // (neg_a, A, neg_b, B, c_mod, C, reuse_a, reuse_b)
    acc = __builtin_amdgcn_wmma_f32_16x16x4_f32(false, A, false, Bm,
                                                (short)0, acc, false, false);

    // D: acc[0] lanes0-15 = row M=0 (x'), acc[1] lanes0-15 = row M=1 (y').
    // Broadcast column (lane>>3) to the 8 lanes handling that pixel.
    const int pixInWave = lane >> 3;               // 0..3
    const float Xs = (wave_bcast(acc[0], pixInWave) + 1.0f) * 256.0f; // *(W/2)
    const float Ys = (wave_bcast(acc[1], pixInWave) + 1.0f) * 256.0f; // *(H/2)

    // ---------------- Bilinear gather + blend ----------------
    int x0 = (int)floorf(Xs);
    int y0 = (int)floorf(Ys);
    int x1 = min(max(x0 + 1, 0), NW - 1);
    int y1 = min(max(y0 + 1, 0), NH - 1);
    x0 = min(max(x0, 0), NW - 1);
    y0 = min(max(y0, 0), NH - 1);
    const float x0f = (float)x0, x1f = (float)x1;
    const float y0f = (float)y0, y1f = (float)y1;
    const float wa = (x1f - Xs) * (y1f - Ys);
    const float wb = (x1f - Xs) * (Ys - y0f);
    const float wc = (Xs - x0f) * (y1f - Ys);
    const float wd = (Xs - x0f) * (Ys - y0f);

    const int rowBase = b * (NH * NW);
    const int ra = rowBase + y0 * NW + x0;   // Ia: (y0,x0)
    const int rb = rowBase + y1 * NW + x0;   // Ib: (y1,x0)
    const int rc = rowBase + y0 * NW + x1;   // Ic: (y0,x1)
    const int rd = rowBase + y1 * NW + x1;   // Id: (y1,x1)
    const int co = chunk * 4;

    // Full 128B pixel vectors fetched as b128 per lane; image stays L2-resident.
    const v4f va = *(const v4f*)(im + (size_t)ra * ND + co);
    const v4f vb = *(const v4f*)(im + (size_t)rb * ND + co);
    const v4f vc = *(const v4f*)(im + (size_t)rc * ND + co);
    const v4f vd = *(const v4f*)(im + (size_t)rd * ND + co);

    v4f r = va * wa + vb * wb + vc * wc + vd * wd;

    // Non-temporal b128 store: 128 MiB output stream must not evict the
    // 128 MiB input image from the 192 MB L2.
    __builtin_nontemporal_store(r, (v4f*)(out + (size_t)pixel * ND + co));
}

extern "C" void kernel_launch(void* const* d_in, const int* in_sizes, int n_in,
                              void* d_out, int out_size, void* d_ws, size_t ws_size,
                              hipStream_t stream) {
    const float* im     = (const float*)d_in[0];
    const float* thetas = (const float*)d_in[1];
    float* out          = (float*)d_out;

    const int totalThreads = NB * NH * NW * 8;   // 8 lanes per pixel
    const int block = 256;                       // 8 wave32 waves / block
    const int grid  = totalThreads / block;      // exact multiple -> EXEC all-ones

    hipLaunchKernelGGL(affine_bilinear_kernel, dim3(grid), dim3(block), 0, stream,
                       im, thetas, out);
}